// BiDirectionalLoss_88364657148111
// MI455X (gfx1250) — compile-verified
//
#include <hip/hip_runtime.h>
#include <hip/hip_bf16.h>
#include <float.h>

#define BATCH 1024
#define DIM 128
#define MARGIN_C 0.5f
#define INTRA_MARGIN_C 0.1f
#define LAMDA_C 0.5f
#define EPS_C 1e-12f
#define NEG_OFFSET_C 100000.0f

typedef float v2f __attribute__((ext_vector_type(2)));
typedef float v8f __attribute__((ext_vector_type(8)));

__global__ void init_out_kernel(float* out) {
    if (threadIdx.x == 0 && blockIdx.x == 0) out[0] = 0.0f;
}

__global__ void row_norms_kernel(const float* __restrict__ feat,
                                 float* __restrict__ norms, int nrows) {
    int i = blockIdx.x * blockDim.x + threadIdx.x;
    if (i >= nrows) return;
    const float4* p = (const float4*)(feat + (size_t)i * DIM);
    float s = 0.0f;
#pragma unroll
    for (int k = 0; k < DIM / 4; ++k) {
        float4 v = p[k];
        s += v.x * v.x + v.y * v.y + v.z * v.z + v.w * v.w;
    }
    norms[i] = s;
}

// One launch computes one "_half": rows of Fa vs columns (= rows) of Fb.
// Workgroup = 16 rows; 4 waves; wave w owns column tiles j0 = (w + 4t)*16.
// GEMM via V_WMMA_F32_16X16X4_F32 with double-buffered B-fragment loads:
// group of 16 K-steps loads in flight while previous group's 16 WMMAs run.
__launch_bounds__(128)
__global__ void half_loss_kernel(const float* __restrict__ Fa,
                                 const float* __restrict__ Fb,
                                 const int* __restrict__ labA,
                                 const int* __restrict__ labB,
                                 const float* __restrict__ nA,
                                 const float* __restrict__ nB,
                                 float* __restrict__ out) {
    __shared__ float s_maxv[16][64];
    __shared__ int   s_maxi[16][64];
    __shared__ float s_minv[16][64];
    __shared__ int   s_mini[16][64];

    const int tid  = threadIdx.x;
    const int wave = tid >> 5;
    const int lane = tid & 31;
    const int m16  = lane & 15;
    const int hi   = lane >> 4;      // which lane-half (0/1)
    const int koff = hi * 2;         // K sub-offset per ISA 16x4 f32 A layout
    const int i0   = blockIdx.x * 16;

    // Preload A fragments for all K: lane holds row (i0+m16), floats at 4k+koff
    v2f afrag[32];
    const float* ap = Fa + (size_t)(i0 + m16) * DIM + koff;
#pragma unroll
    for (int k = 0; k < 32; ++k) afrag[k] = *(const v2f*)(ap + 4 * k);

    // Per-lane row metadata: this lane's accumulators cover rows v + 8*hi
    int   rlab[8];
    float rnrm[8];
#pragma unroll
    for (int v = 0; v < 8; ++v) {
        int row = i0 + v + 8 * hi;
        rlab[v] = labA[row];
        rnrm[v] = nA[row];
    }

    float bestPos[8]; int bestPosIdx[8];
    float bestNeg[8]; int bestNegIdx[8];
#pragma unroll
    for (int v = 0; v < 8; ++v) {
        bestPos[v] = 0.0f;     bestPosIdx[v] = 0;
        bestNeg[v] = FLT_MAX;  bestNegIdx[v] = 0;
    }

    // B row base for tile t: Fb + ((w+4t)*16 + m16)*DIM + koff.
    // Tile stride along t is 64*DIM floats.
    const float* bp0 = Fb + (size_t)(wave * 16 + m16) * DIM + koff;

    // Double-buffered B fragments: two groups of 16 K-steps.
    v2f bf0[16];
    v2f bf1[16];

    // Prologue: load group 0 of tile 0.
#pragma unroll
    for (int kk = 0; kk < 16; ++kk) bf0[kk] = *(const v2f*)(bp0 + 4 * kk);

    for (int t = 0; t < 16; ++t) {
        const float* bp     = bp0 + (size_t)t * 64 * DIM;
        const float* bpnext = bp + (size_t)64 * DIM;
        v8f acc = {};

        // ---- group 0: prefetch group 1, compute k = 0..15 from bf0 ----
#pragma unroll
        for (int kk = 0; kk < 16; ++kk)
            bf1[kk] = *(const v2f*)(bp + 4 * (16 + kk));
#pragma unroll
        for (int kk = 0; kk < 16; ++kk)
            acc = __builtin_amdgcn_wmma_f32_16x16x4_f32(
                false, afrag[kk], false, bf0[kk], (short)0, acc, false, false);

        // ---- group 1: prefetch next tile's group 0, compute k = 16..31 ----
        if (t < 15) {
#pragma unroll
            for (int kk = 0; kk < 16; ++kk)
                bf0[kk] = *(const v2f*)(bpnext + 4 * kk);
        }
#pragma unroll
        for (int kk = 0; kk < 16; ++kk)
            acc = __builtin_amdgcn_wmma_f32_16x16x4_f32(
                false, afrag[16 + kk], false, bf1[kk], (short)0, acc, false, false);

        // Epilogue: lane holds column (j0+m16) for rows v+8*hi (VGPR v)
        const int   j0  = (wave + 4 * t) * 16;
        const int   col = j0 + m16;
        const int   lb  = labB[col];
        const float nb  = nB[col];
#pragma unroll
        for (int v = 0; v < 8; ++v) {
            float d2  = rnrm[v] + nb - 2.0f * acc[v];
            float dis = sqrtf(fmaxf(d2, 0.0f) + EPS_C);
            bool  s   = (rlab[v] == lb);
            float pv  = s ? dis : 0.0f;                 // dis * s
            if (pv > bestPos[v]) { bestPos[v] = pv; bestPosIdx[v] = col; }
            float nv  = s ? (dis + NEG_OFFSET_C) : dis; // dis + NEG_OFFSET*s
            if (nv < bestNeg[v]) { bestNeg[v] = nv; bestNegIdx[v] = col; }
        }
    }

    // Cross-lane / cross-wave reduction via LDS
    const int prt = wave * 16 + m16;  // 0..63 partial slot
#pragma unroll
    for (int v = 0; v < 8; ++v) {
        int row = v + 8 * hi;  // local row 0..15
        s_maxv[row][prt] = bestPos[v]; s_maxi[row][prt] = bestPosIdx[v];
        s_minv[row][prt] = bestNeg[v]; s_mini[row][prt] = bestNegIdx[v];
    }
    __syncthreads();

    if (tid < 16) {
        float fp = 0.0f;    int pp = 0;
        float cn = FLT_MAX; int pn = 0;
        for (int p = 0; p < 64; ++p) {
            float mv = s_maxv[tid][p]; int mi = s_maxi[tid][p];
            if (mv > fp || (mv == fp && mi < pp)) { fp = mv; pp = mi; }
            float nv = s_minv[tid][p]; int ni = s_mini[tid][p];
            if (nv < cn || (nv == cn && ni < pn)) { cn = nv; pn = ni; }
        }
        float cross = fmaxf(fp - cn + MARGIN_C, 0.0f);

        // intra term: gathered = || Fb[pp] - Fb[pn] ||  (f_intra == f_b)
        float sum2 = 0.0f;
        const float4* r1 = (const float4*)(Fb + (size_t)pp * DIM);
        const float4* r2 = (const float4*)(Fb + (size_t)pn * DIM);
#pragma unroll
        for (int k = 0; k < DIM / 4; ++k) {
            float4 x = r1[k], y = r2[k];
            float dx = x.x - y.x, dy = x.y - y.y, dz = x.z - y.z, dw = x.w - y.w;
            sum2 += dx * dx + dy * dy + dz * dz + dw * dw;
        }
        float gathered = sqrtf(sum2 + EPS_C);
        float intra = fmaxf(INTRA_MARGIN_C - gathered, 0.0f);

        atomicAdd(out, (cross + LAMDA_C * intra) * (1.0f / (float)BATCH));
    }
}

extern "C" void kernel_launch(void* const* d_in, const int* in_sizes, int n_in,
                              void* d_out, int out_size, void* d_ws, size_t ws_size,
                              hipStream_t stream) {
    const float* feat   = (const float*)d_in[0];
    const int*   label1 = (const int*)d_in[1];
    const int*   label2 = (const int*)d_in[2];
    float* out = (float*)d_out;

    float* norms = (float*)d_ws;  // 2048 floats of scratch
    const float* F1 = feat;
    const float* F2 = feat + (size_t)BATCH * DIM;
    const float* n1 = norms;
    const float* n2 = norms + BATCH;

    init_out_kernel<<<1, 64, 0, stream>>>(out);
    row_norms_kernel<<<(2 * BATCH + 255) / 256, 256, 0, stream>>>(feat, norms, 2 * BATCH);
    half_loss_kernel<<<BATCH / 16, 128, 0, stream>>>(F1, F2, label1, label2, n1, n2, out);
    half_loss_kernel<<<BATCH / 16, 128, 0, stream>>>(F2, F1, label2, label1, n2, n1, out);
}